// EnhancedDragGNN_52493090292285
// MI455X (gfx1250) — compile-verified
//
#include <hip/hip_runtime.h>
#include <math.h>

// ---------------------------------------------------------------------------
// EnhancedDragGNN forward for MI455X (gfx1250, wave32).
// Dense GEMMs use V_WMMA_F32_16X16X4_F32, one 32x32 macro-tile per wave
// (2x2 WMMA tiles, 4 accumulators) with branch-free b64/b32 loads.
// Edge segment ops use float4 gathers + global f32 atomics (L2-resident).
// Requires N % 32 == 0 (N_NODES = 100000 = 32 * 3125). K padded to mult of 4.
// Workspace requirement: ~520 MB (jk[N,768] + 2x tmp[N,256] + small).
// ---------------------------------------------------------------------------

typedef __attribute__((ext_vector_type(2))) float v2f;
typedef __attribute__((ext_vector_type(8))) float v8f;

static inline int cdiv_i(long long a, long long b) { return (int)((a + b - 1) / b); }

// ---------------------------- small utilities ------------------------------

__global__ void k_fill(float* __restrict__ p, float v, int n) {
  int i = blockIdx.x * blockDim.x + threadIdx.x;
  if (i < n) p[i] = v;
}

__global__ void k_deg(const int* __restrict__ dstI, float* __restrict__ deg, int E) {
  int e = blockIdx.x * blockDim.x + threadIdx.x;
  if (e < E) atomicAdd(&deg[dstI[e]], 1.0f);
}

__global__ void k_rsqrt_inplace(float* __restrict__ p, int n) {
  int i = blockIdx.x * blockDim.x + threadIdx.x;
  if (i < n) p[i] = rsqrtf(p[i]);
}

// pad [N,3] -> [N,4] (zero 4th column) so the GEMM K-loop is branch-free
__global__ void k_pad_x(const float* __restrict__ x, float* __restrict__ xp, int N) {
  int n = blockIdx.x * blockDim.x + threadIdx.x;
  if (n < N) {
    int i3 = n * 3, i4 = n * 4;
    xp[i4 + 0] = x[i3 + 0];
    xp[i4 + 1] = x[i3 + 1];
    xp[i4 + 2] = x[i3 + 2];
    xp[i4 + 3] = 0.0f;
  }
}

__global__ void k_bias_relu(float* __restrict__ x, const float* __restrict__ b,
                            int total, int F) {
  int i = blockIdx.x * blockDim.x + threadIdx.x;
  if (i < total) {
    float v = x[i] + b[i % F];
    x[i] = v > 0.0f ? v : 0.0f;
  }
}

// ------------------------- WMMA fp32 GEMM: Y = X @ W -----------------------
// X: [N,K] row-major, leading dim ldX (allows strided jk slices).
// W: [K,F] row-major. Y: [N,F] row-major.
// One wave computes a 32x32 macro-tile = 2x2 WMMA 16x16 tiles.
// Preconditions: N % 32 == 0, F % 32 == 0, K % 4 == 0.
// A lane layout (16x4 f32): lane L, vgpr j -> A[M=L%16][K = 2*(L/16)+j]
// B lane layout (4x16 f32): lane L, vgpr j -> B[K = 2*(L/16)+j][N=L%16]
// C/D layout (16x16 f32):   vgpr v, lane L -> C[M = v + 8*(L/16)][N=L%16]

__global__ void k_gemm_f32wmma(const float* __restrict__ X, int ldX,
                               const float* __restrict__ W,
                               float* __restrict__ Y,
                               int N, int K, int F) {
  const int wave = blockIdx.x * (blockDim.x >> 5) + (threadIdx.x >> 5);
  const int lane = threadIdx.x & 31;
  const int colMT = F >> 5;
  const int rowMT = N >> 5;
  if (wave >= rowMT * colMT) return;            // wave-uniform exit

  const int rt = wave / colMT;
  const int ct = wave - rt * colMT;
  const int m  = lane & 15;
  const int kh = lane >> 4;                     // K half (0/1)
  const int k0 = kh << 1;

  const float* xp0 = X + ((rt << 5) + m) * ldX + k0;      // rows rt*32 + m
  const float* xp1 = xp0 + (ldX << 4);                    // rows rt*32 + 16 + m
  const float* wp0 = W + k0 * F + (ct << 5) + m;          // cols ct*32 + m
  const float* wp1 = wp0 + 16;                            // cols ct*32 + 16 + m
  const int wStep = F << 2;                               // 4 K rows

  v8f acc00 = {}, acc01 = {}, acc10 = {}, acc11 = {};
  for (int kk = 0; kk < K; kk += 4) {
    v2f a0 = *(const v2f*)xp0;
    v2f a1 = *(const v2f*)xp1;
    v2f b0, b1;
    b0.x = wp0[0]; b0.y = wp0[F];
    b1.x = wp1[0]; b1.y = wp1[F];
    xp0 += 4; xp1 += 4; wp0 += wStep; wp1 += wStep;
    acc00 = __builtin_amdgcn_wmma_f32_16x16x4_f32(false, a0, false, b0, (short)0, acc00, false, false);
    acc01 = __builtin_amdgcn_wmma_f32_16x16x4_f32(false, a0, false, b1, (short)0, acc01, false, false);
    acc10 = __builtin_amdgcn_wmma_f32_16x16x4_f32(false, a1, false, b0, (short)0, acc10, false, false);
    acc11 = __builtin_amdgcn_wmma_f32_16x16x4_f32(false, a1, false, b1, (short)0, acc11, false, false);
  }

  float* y0 = Y + ((rt << 5) + (kh << 3)) * F + (ct << 5) + m;   // rows +kh*8
#pragma unroll
  for (int v = 0; v < 8; ++v) {
    y0[v * F]              = acc00[v];
    y0[v * F + 16]         = acc01[v];
    y0[(v + 16) * F]       = acc10[v];
    y0[(v + 16) * F + 16]  = acc11[v];
  }
}

// --------------------------- GCN edge scatter ------------------------------
// out[dst] += xw[src] * dinv[src]*dinv[dst], over E real edges + N self loops.
__global__ void k_gcn_scatter(const int* __restrict__ srcI, const int* __restrict__ dstI,
                              const float* __restrict__ xw, const float* __restrict__ dinv,
                              float* __restrict__ out, int E, int N, int F) {
  const int F4 = F >> 2;
  int t = blockIdx.x * blockDim.x + threadIdx.x;
  int total = (E + N) * F4;
  if (t >= total) return;
  int e = t / F4;
  int f = (t - e * F4) << 2;
  int s = (e < E) ? srcI[e] : (e - E);
  int d = (e < E) ? dstI[e] : (e - E);
  float w = dinv[s] * dinv[d];
  const float4 v = *(const float4*)(xw + s * F + f);
  float* o = out + d * F + f;
  atomicAdd(o + 0, v.x * w);
  atomicAdd(o + 1, v.y * w);
  atomicAdd(o + 2, v.z * w);
  atomicAdd(o + 3, v.w * w);
}

// ------------------------------- BatchNorm ---------------------------------

__global__ void k_bn_stats(const float* __restrict__ x, int ldX,
                           float* __restrict__ sums, int N, int F) {
  int c = threadIdx.x;                      // blockDim.x == F
  float s = 0.0f, s2 = 0.0f;
  for (int r = blockIdx.x; r < N; r += gridDim.x) {
    float v = x[r * ldX + c];
    s += v; s2 += v * v;
  }
  atomicAdd(&sums[c], s);
  atomicAdd(&sums[F + c], s2);
}

__global__ void k_bn_finalize(const float* __restrict__ sums,
                              const float* __restrict__ g, const float* __restrict__ b,
                              float* __restrict__ ss, int F, float invN) {
  int c = threadIdx.x;
  if (c < F) {
    float mu  = sums[c] * invN;
    float var = sums[F + c] * invN - mu * mu;
    float sc  = g[c] * rsqrtf(var + 1e-5f);
    ss[c]     = sc;
    ss[F + c] = b[c] - mu * sc;
  }
}

__global__ void k_bn_apply_relu(const float* __restrict__ x, int ldX,
                                const float* __restrict__ ss,
                                float* __restrict__ y, int ldY, int N, int F) {
  int t = blockIdx.x * blockDim.x + threadIdx.x;
  if (t >= N * F) return;
  int r = t / F, c = t - r * F;
  float v = x[r * ldX + c] * ss[c] + ss[F + c];
  y[r * ldY + c] = v > 0.0f ? v : 0.0f;
}

// --------------------------------- GAT -------------------------------------

// attention logits per node: al_s[n,h] = <xw[n,h,:], a_src[h,:]> (wave32 reduce)
__global__ void k_gat_al(const float* __restrict__ xw,
                         const float* __restrict__ as, const float* __restrict__ ad,
                         float* __restrict__ al_s, float* __restrict__ al_d,
                         int N, int H, int F) {
  int wave = blockIdx.x * (blockDim.x >> 5) + (threadIdx.x >> 5);
  int lane = threadIdx.x & 31;
  if (wave >= N * H) return;
  int n = wave / H, h = wave - n * H;
  float ssum = 0.0f, dsum = 0.0f;
  for (int f = lane; f < F; f += 32) {
    float v = xw[n * 256 + h * F + f];      // H*F == 256 for both GAT layers
    ssum += v * as[h * F + f];
    dsum += v * ad[h * F + f];
  }
  for (int o = 16; o > 0; o >>= 1) {
    ssum += __shfl_xor(ssum, o, 32);
    dsum += __shfl_xor(dsum, o, 32);
  }
  if (lane == 0) { al_s[n * H + h] = ssum; al_d[n * H + h] = dsum; }
}

__device__ inline float atomicMaxF(float* addr, float val) {
  if (val >= 0.0f)
    return __int_as_float(atomicMax((int*)addr, __float_as_int(val)));
  else
    return __uint_as_float(atomicMin((unsigned int*)addr, __float_as_uint(val)));
}

__global__ void k_gat_max(const int* __restrict__ srcI, const int* __restrict__ dstI,
                          const float* __restrict__ al_s, const float* __restrict__ al_d,
                          float* __restrict__ mbuf, int E, int N, int H) {
  int t = blockIdx.x * blockDim.x + threadIdx.x;
  int total = (E + N) * H;
  if (t >= total) return;
  int e = t / H, h = t - e * H;
  int s = (e < E) ? srcI[e] : (e - E);
  int d = (e < E) ? dstI[e] : (e - E);
  float ev = al_s[s * H + h] + al_d[d * H + h];
  ev = ev > 0.0f ? ev : 0.2f * ev;                 // leaky_relu(., 0.2)
  atomicMaxF(&mbuf[d * H + h], ev);
}

// one 256-thread block per edge; H*F == 256 for both GAT layers
__global__ void k_gat_scatter(const int* __restrict__ srcI, const int* __restrict__ dstI,
                              const float* __restrict__ al_s, const float* __restrict__ al_d,
                              const float* __restrict__ mbuf, const float* __restrict__ xw,
                              float* __restrict__ den, float* __restrict__ num,
                              int E, int N, int H, int F) {
  int e = blockIdx.x;
  int t = threadIdx.x;                 // 0..255 == h*F + f
  int h = t / F, f = t - h * F;
  int s = (e < E) ? srcI[e] : (e - E);
  int d = (e < E) ? dstI[e] : (e - E);
  float ev = al_s[s * H + h] + al_d[d * H + h];
  ev = ev > 0.0f ? ev : 0.2f * ev;
  float ex = expf(ev - mbuf[d * H + h]);
  atomicAdd(&num[d * 256 + t], ex * xw[s * 256 + t]);
  if (f == 0) atomicAdd(&den[d * H + h], ex);
}

__global__ void k_gat_final(const float* __restrict__ num, const float* __restrict__ den,
                            const float* __restrict__ bias, float* __restrict__ y,
                            int ldY, int N, int H, int F) {
  int t = blockIdx.x * blockDim.x + threadIdx.x;
  if (t >= N * 256) return;
  int n = t / 256, c = t - n * 256;
  int h = c / F;
  y[n * ldY + c] = num[n * 256 + c] / den[n * H + h] + bias[c];
}

// ------------------------------- Pooling -----------------------------------

__global__ void k_counts(const int* __restrict__ batch, float* __restrict__ counts, int N) {
  int n = blockIdx.x * blockDim.x + threadIdx.x;
  if (n < N) atomicAdd(&counts[batch[n]], 1.0f);
}

__global__ void k_pool(const float* __restrict__ jk, const int* __restrict__ batch,
                       float* __restrict__ pooled, int N) {
  int t = blockIdx.x * blockDim.x + threadIdx.x;
  if (t >= N * 768) return;
  int n = t / 768, c = t - n * 768;
  atomicAdd(&pooled[batch[n] * 768 + c], jk[n * 768 + c]);
}

// ----------------------------- MLP head ------------------------------------
// pooled[8,768] -> fc1(768,128)+relu -> BN over 8 graphs -> fc2(128,1).
__global__ void k_head(const float* __restrict__ pooled, const float* __restrict__ counts,
                       const float* __restrict__ W1, const float* __restrict__ b1,
                       const float* __restrict__ g1, const float* __restrict__ be1,
                       const float* __restrict__ W2, const float* __restrict__ b2,
                       float* __restrict__ out) {
  __shared__ float red[128];
  int c = threadIdx.x;                  // 0..127
  float h[8];
  for (int gi = 0; gi < 8; ++gi) {
    float inv = 1.0f / counts[gi];
    float acc = b1[c];
    for (int k = 0; k < 768; ++k)
      acc += pooled[gi * 768 + k] * inv * W1[k * 128 + c];
    h[gi] = acc > 0.0f ? acc : 0.0f;
  }
  float mu = 0.0f;
  for (int gi = 0; gi < 8; ++gi) mu += h[gi];
  mu *= 0.125f;
  float var = 0.0f;
  for (int gi = 0; gi < 8; ++gi) { float d = h[gi] - mu; var += d * d; }
  var *= 0.125f;
  float sc = g1[c] * rsqrtf(var + 1e-5f);
  float w2 = W2[c];
  for (int gi = 0; gi < 8; ++gi) {
    float hb = (h[gi] - mu) * sc + be1[c];
    red[c] = hb * w2;
    __syncthreads();
    for (int off = 64; off > 0; off >>= 1) {
      if (c < off) red[c] += red[c + off];
      __syncthreads();
    }
    if (c == 0) out[gi] = red[0] + b2[0];
    __syncthreads();
  }
}

// ------------------------------ orchestration ------------------------------

extern "C" void kernel_launch(void* const* d_in, const int* in_sizes, int n_in,
                              void* d_out, int out_size, void* d_ws, size_t ws_size,
                              hipStream_t stream) {
  const int N = in_sizes[0] / 3;
  const int E = in_sizes[1] / 2;
  const int Etot = E + N;

  const float* x     = (const float*)d_in[0];
  const int*   ei    = (const int*)d_in[1];
  const int*   batch = (const int*)d_in[2];
  const int* srcI = ei;       // edge_index[0]
  const int* dstI = ei + E;   // edge_index[1]

  const float* gcn1_W = (const float*)d_in[3];
  const float* gcn1_b = (const float*)d_in[4];
  const float* gat1_W = (const float*)d_in[5];
  const float* gat1_as= (const float*)d_in[6];
  const float* gat1_ad= (const float*)d_in[7];
  const float* gat1_b = (const float*)d_in[8];
  const float* bn1_g  = (const float*)d_in[9];
  const float* bn1_b  = (const float*)d_in[10];
  const float* gcn2_W = (const float*)d_in[11];
  const float* gcn2_b = (const float*)d_in[12]; (void)gcn2_b;  // folded: BN removes channel bias
  const float* gat2_W = (const float*)d_in[13];
  const float* gat2_as= (const float*)d_in[14];
  const float* gat2_ad= (const float*)d_in[15];
  const float* gat2_b = (const float*)d_in[16];
  const float* bn2_g  = (const float*)d_in[17];
  const float* bn2_b  = (const float*)d_in[18];
  const float* gcn3_W = (const float*)d_in[19];
  const float* gcn3_b = (const float*)d_in[20]; (void)gcn3_b;  // folded: BN removes channel bias
  const float* fc1_W  = (const float*)d_in[21];
  const float* fc1_b  = (const float*)d_in[22];
  const float* fc1_g  = (const float*)d_in[23];
  const float* fc1_be = (const float*)d_in[24];
  const float* fc2_W  = (const float*)d_in[25];
  const float* fc2_b  = (const float*)d_in[26];

  // workspace layout (floats)
  float* jk     = (float*)d_ws;              // [N,768] : x1|x2|x3
  float* tmpA   = jk + (long long)N * 768;   // [N,256] GEMM outputs
  float* tmpB   = tmpA + (long long)N * 256; // [N,256] aggregation target / staging
  float* dinv   = tmpB + (long long)N * 256; // [N]
  float* al_s   = dinv + N;                  // [N,4]
  float* al_d   = al_s + (long long)N * 4;   // [N,4]
  float* mbuf   = al_d + (long long)N * 4;   // [N,4]
  float* den    = mbuf + (long long)N * 4;   // [N,4]
  float* sums   = den  + (long long)N * 4;   // [512]
  float* ssbuf  = sums + 512;                // [512]
  float* pooled = ssbuf + 512;               // [8,768]
  float* counts = pooled + 8 * 768;          // [8]
  float* out    = (float*)d_out;

  const int T = 256;
  const int WPB = T / 32;                    // 8 waves/block

  auto gemm = [&](const float* X, int ldX, const float* W, float* Y, int K, int F) {
    int tiles = (N >> 5) * (F >> 5);         // 32x32 macro-tiles
    k_gemm_f32wmma<<<cdiv_i(tiles, WPB), T, 0, stream>>>(X, ldX, W, Y, N, K, F);
  };

  // ---- degree / symmetric norm -------------------------------------------
  k_fill<<<cdiv_i(N, T), T, 0, stream>>>(dinv, 1.0f, N);       // self loops
  k_deg<<<cdiv_i(E, T), T, 0, stream>>>(dstI, dinv, E);
  k_rsqrt_inplace<<<cdiv_i(N, T), T, 0, stream>>>(dinv, N);

  // ---- GCN1: [N,3]->pad[N,4] -> [N,64], +bias, relu ----------------------
  k_pad_x<<<cdiv_i(N, T), T, 0, stream>>>(x, tmpB, N);         // stage padded X in tmpB
  gemm(tmpB, 4, gcn1_W, tmpA, 4, 64);
  hipMemsetAsync(tmpB, 0, (size_t)N * 64 * sizeof(float), stream);
  k_gcn_scatter<<<cdiv_i((long long)Etot * 16, T), T, 0, stream>>>(srcI, dstI, tmpA, dinv, tmpB, E, N, 64);
  k_bias_relu<<<cdiv_i((long long)N * 64, T), T, 0, stream>>>(tmpB, gcn1_b, N * 64, 64);

  // ---- GAT1: [N,64] -> [N,256] (H=4,F=64) -> jk[:,0:256] ------------------
  gemm(tmpB, 64, gat1_W, tmpA, 64, 256);
  hipMemsetAsync(tmpB, 0, (size_t)N * 256 * sizeof(float), stream);
  hipMemsetAsync(den, 0, (size_t)N * 4 * sizeof(float), stream);
  k_fill<<<cdiv_i((long long)N * 4, T), T, 0, stream>>>(mbuf, -1e30f, N * 4);
  k_gat_al<<<cdiv_i((long long)N * 4, WPB), T, 0, stream>>>(tmpA, gat1_as, gat1_ad, al_s, al_d, N, 4, 64);
  k_gat_max<<<cdiv_i((long long)Etot * 4, T), T, 0, stream>>>(srcI, dstI, al_s, al_d, mbuf, E, N, 4);
  k_gat_scatter<<<Etot, 256, 0, stream>>>(srcI, dstI, al_s, al_d, mbuf, tmpA, den, tmpB, E, N, 4, 64);
  k_gat_final<<<cdiv_i((long long)N * 256, T), T, 0, stream>>>(tmpB, den, gat1_b, jk + 0, 768, N, 4, 64);

  // ---- GCN2: [N,256] -> [N,128], BN+relu (bias folded into BN) ------------
  gemm(jk + 0, 768, gcn2_W, tmpA, 256, 128);
  hipMemsetAsync(tmpB, 0, (size_t)N * 128 * sizeof(float), stream);
  k_gcn_scatter<<<cdiv_i((long long)Etot * 32, T), T, 0, stream>>>(srcI, dstI, tmpA, dinv, tmpB, E, N, 128);
  hipMemsetAsync(sums, 0, 2 * 128 * sizeof(float), stream);
  k_bn_stats<<<512, 128, 0, stream>>>(tmpB, 128, sums, N, 128);
  k_bn_finalize<<<1, 128, 0, stream>>>(sums, bn1_g, bn1_b, ssbuf, 128, 1.0f / (float)N);
  k_bn_apply_relu<<<cdiv_i((long long)N * 128, T), T, 0, stream>>>(tmpB, 128, ssbuf, tmpB, 128, N, 128);

  // ---- GAT2: [N,128] -> [N,256] (H=2,F=128) -> jk[:,256:512] --------------
  gemm(tmpB, 128, gat2_W, tmpA, 128, 256);
  hipMemsetAsync(tmpB, 0, (size_t)N * 256 * sizeof(float), stream);
  hipMemsetAsync(den, 0, (size_t)N * 2 * sizeof(float), stream);
  k_fill<<<cdiv_i((long long)N * 2, T), T, 0, stream>>>(mbuf, -1e30f, N * 2);
  k_gat_al<<<cdiv_i((long long)N * 2, WPB), T, 0, stream>>>(tmpA, gat2_as, gat2_ad, al_s, al_d, N, 2, 128);
  k_gat_max<<<cdiv_i((long long)Etot * 2, T), T, 0, stream>>>(srcI, dstI, al_s, al_d, mbuf, E, N, 2);
  k_gat_scatter<<<Etot, 256, 0, stream>>>(srcI, dstI, al_s, al_d, mbuf, tmpA, den, tmpB, E, N, 2, 128);
  k_gat_final<<<cdiv_i((long long)N * 256, T), T, 0, stream>>>(tmpB, den, gat2_b, jk + 256, 768, N, 2, 128);

  // ---- GCN3: [N,256] -> [N,256], BN+relu -> jk[:,512:768] -----------------
  gemm(jk + 256, 768, gcn3_W, tmpA, 256, 256);
  hipMemsetAsync(tmpB, 0, (size_t)N * 256 * sizeof(float), stream);
  k_gcn_scatter<<<cdiv_i((long long)Etot * 64, T), T, 0, stream>>>(srcI, dstI, tmpA, dinv, tmpB, E, N, 256);
  hipMemsetAsync(sums, 0, 2 * 256 * sizeof(float), stream);
  k_bn_stats<<<512, 256, 0, stream>>>(tmpB, 256, sums, N, 256);
  k_bn_finalize<<<1, 256, 0, stream>>>(sums, bn2_g, bn2_b, ssbuf, 256, 1.0f / (float)N);
  k_bn_apply_relu<<<cdiv_i((long long)N * 256, T), T, 0, stream>>>(tmpB, 256, ssbuf, jk + 512, 768, N, 256);

  // ---- mean pool + head ---------------------------------------------------
  hipMemsetAsync(pooled, 0, (8 * 768 + 8) * sizeof(float), stream);  // pooled+counts
  k_counts<<<cdiv_i(N, T), T, 0, stream>>>(batch, counts, N);
  k_pool<<<cdiv_i((long long)N * 768, T), T, 0, stream>>>(jk, batch, pooled, N);
  k_head<<<1, 128, 0, stream>>>(pooled, counts, fc1_W, fc1_b, fc1_g, fc1_be, fc2_W, fc2_b, out);
}